// RecNet_39745627357206
// MI455X (gfx1250) — compile-verified
//
#include <hip/hip_runtime.h>

typedef float v2f __attribute__((ext_vector_type(2)));
typedef float v8f __attribute__((ext_vector_type(8)));
typedef unsigned int u32x4 __attribute__((ext_vector_type(4)));
typedef int i32x4 __attribute__((ext_vector_type(4)));
typedef int i32x8 __attribute__((ext_vector_type(8)));

#if __has_builtin(__builtin_amdgcn_tensor_load_to_lds) && \
    __has_builtin(__builtin_amdgcn_s_wait_tensorcnt)
#define HAVE_TDM 1
#else
#define HAVE_TDM 0
#endif

#define KC   32     // K-chunk staged per double-buffer slot
#define LDP  34     // padded LDS row stride (floats): 16 consecutive rows -> 16 distinct banks

#if HAVE_TDM
// Issue one TDM 2D-tile load: rows x KC f32 tile, global row stride = k_stride
// elements, deposited into LDS with a 2-DWORD pad after every 32 DWORDs
// (so LDS row stride becomes LDP=34 floats). D# packing per CDNA5 ISA 8.3/8.4.
__device__ __forceinline__ void tdm_issue_2d(const float* gsrc, const void* lds_dst,
                                             int rows, int k_stride)
{
    unsigned long long ga = (unsigned long long)(uintptr_t)gsrc;
    u32x4 g0;
    g0[0] = 1u;                                              // count=1, user descriptor
    g0[1] = (unsigned)(uintptr_t)lds_dst;                    // lds_addr (bytes)
    g0[2] = (unsigned)ga;                                    // global_addr[31:0]
    g0[3] = ((unsigned)(ga >> 32) & 0x01FFFFFFu)             // global_addr[56:32]
            | 0x80000000u;                                   // type=2 ("image")
    i32x8 g1;
    g1[0] = (int)((2u << 16)      // data_size = 4 bytes
                | (1u << 20)      // pad_enable
                | (4u << 22)      // pad_interval: every 32 DWORDs
                | (1u << 25));    // pad_amount: 2 DWORDs  -> LDS stride 34 floats
    g1[1] = 0;                    // atomic_barrier=0; tensor_dim0 lo16 (td0=0x40000000 -> 0)
    g1[2] = 0x00004000;           // tensor_dim0 hi16 = 0x4000; tensor_dim1 lo16 = 0
    g1[3] = (int)(0x00004000u | ((unsigned)KC << 16));   // tensor_dim1 hi16; tile_dim0 = KC
    g1[4] = rows;                 // tile_dim1 = rows; tile_dim2 = 0
    g1[5] = k_stride;             // tensor_dim0_stride lo32 (elements)
    g1[6] = 0;                    // stride hi16; tensor_dim1_stride lo16
    g1[7] = 0;
    i32x4 z4 = {0, 0, 0, 0};
#if __has_include(<hip/amd_detail/amd_gfx1250_TDM.h>)
    i32x8 z8 = {0, 0, 0, 0, 0, 0, 0, 0};
    __builtin_amdgcn_tensor_load_to_lds(g0, g1, z4, z4, z8, 0);   // clang-23 / therock
#else
    __builtin_amdgcn_tensor_load_to_lds(g0, g1, z4, z4, 0);       // ROCm 7.2 / clang-22
#endif
}
#else
// Fallback: blocking cooperative global->LDS copy (all 256 threads).
__device__ __forceinline__ void stage_copy(const float* gsrc, float* lds_dst,
                                           int rows, int k_stride, int tid)
{
    const int vecs_per_row = KC / 4;
    for (int i = tid; i < rows * vecs_per_row; i += 256) {
        const int r = i / vecs_per_row;
        const int c = (i - r * vecs_per_row) * 4;
        const float4 v = *(const float4*)(gsrc + (size_t)r * k_stride + c);
        float* d = lds_dst + r * LDP + c;
        d[0] = v.x; d[1] = v.y; d[2] = v.z; d[3] = v.w;
    }
}
#endif

// Fused tile kernel:
//   C[m,n] = act( (addX ? X[m,n] : 0) + bias[n] + (useA ? sum_k A[m,k]*W[n,k] : 0) )
// A:[M,K], W:[N,K] row-major (matmul = A @ W^T), X,C:[M,N].
// Workgroup = 256 threads (8 waves), macro-tile 64(M) x 128(N); wave tile 16x64
// (4 x 16x16 WMMA accumulators sharing the A fragment). K is pipelined through
// LDS in KC=32 double-buffered chunks fed by the Tensor Data Mover, with the
// TDM pad feature giving a conflict-free 34-float LDS row stride.
__global__ __launch_bounds__(256)
void rnn_gemm_wmma_f32(const float* __restrict__ A,
                       const float* __restrict__ W,
                       const float* __restrict__ X,
                       const float* __restrict__ bias,
                       float* __restrict__ C,
                       int M, int N, int K,
                       int useA, int addX, int relu)
{
    __shared__ float As[2][64  * LDP];
    __shared__ float Bs[2][128 * LDP];

    const int tid   = threadIdx.x;
    const int lane  = tid & 31;
    const int wid   = tid >> 5;
    const int waveM = wid >> 1;          // 0..3  -> 16-row slice of the 64-row tile
    const int waveN = wid & 1;           // 0..1  -> 64-col slice of the 128-col tile
    const int half  = lane >> 4;         // f32 WMMA: K pair select (K+{0,1} vs K+{2,3})
    const int l15   = lane & 15;

    const int blockM = blockIdx.y * 64;
    const int blockN = blockIdx.x * 128;

    v8f acc0 = {}; v8f acc1 = {}; v8f acc2 = {}; v8f acc3 = {};

    if (useA) {
        const int nchunks = K / KC;
        const float* Ag = A + (size_t)blockM * K;
        const float* Wg = W + (size_t)blockN * K;

#if HAVE_TDM
        if (tid < 32) {                      // wave 0 drives the TDM
            tdm_issue_2d(Ag, &As[0][0], 64, K);
            tdm_issue_2d(Wg, &Bs[0][0], 128, K);
        }
#else
        stage_copy(Ag, &As[0][0], 64, K, tid);
        stage_copy(Wg, &Bs[0][0], 128, K, tid);
#endif

        const int arow = waveM * 16 + l15;
        const int brow = waveN * 64 + l15;

        for (int c = 0; c < nchunks; ++c) {
            const int cur = c & 1;

#if HAVE_TDM
            if (tid < 32) {
                if (c + 1 < nchunks) {       // prefetch next chunk into the other buffer
                    const int k1 = (c + 1) * KC;
                    tdm_issue_2d(Ag + k1, &As[cur ^ 1][0], 64, K);
                    tdm_issue_2d(Wg + k1, &Bs[cur ^ 1][0], 128, K);
                    __builtin_amdgcn_s_wait_tensorcnt(2);   // chunk c landed; c+1 in flight
                } else {
                    __builtin_amdgcn_s_wait_tensorcnt(0);
                }
            }
#else
            if (c + 1 < nchunks) {
                const int k1 = (c + 1) * KC;
                stage_copy(Ag + k1, &As[cur ^ 1][0], 64, K, tid);
                stage_copy(Wg + k1, &Bs[cur ^ 1][0], 128, K, tid);
            }
#endif
            __syncthreads();                 // chunk c visible to all waves

            #pragma unroll
            for (int kk = 0; kk < KC; kk += 4) {
                const int ko = kk + 2 * half;
                v2f a  = *(const v2f*)&As[cur][arow * LDP + ko];
                v2f b0 = *(const v2f*)&Bs[cur][(brow     ) * LDP + ko];
                v2f b1 = *(const v2f*)&Bs[cur][(brow + 16) * LDP + ko];
                v2f b2 = *(const v2f*)&Bs[cur][(brow + 32) * LDP + ko];
                v2f b3 = *(const v2f*)&Bs[cur][(brow + 48) * LDP + ko];
                acc0 = __builtin_amdgcn_wmma_f32_16x16x4_f32(false, a, false, b0,
                                                             (short)0, acc0, false, false);
                acc1 = __builtin_amdgcn_wmma_f32_16x16x4_f32(false, a, false, b1,
                                                             (short)0, acc1, false, false);
                acc2 = __builtin_amdgcn_wmma_f32_16x16x4_f32(false, a, false, b2,
                                                             (short)0, acc2, false, false);
                acc3 = __builtin_amdgcn_wmma_f32_16x16x4_f32(false, a, false, b3,
                                                             (short)0, acc3, false, false);
            }
            __syncthreads();                 // done reading buf cur before it is re-filled
        }
    }

    // Epilogue: bias + optional X add + optional ReLU.
    // f32 WMMA C/D layout: VGPR v, lane half h -> row v + 8h, col (lane&15).
    const int m0 = blockM + waveM * 16 + (half << 3);
    #pragma unroll
    for (int j = 0; j < 4; ++j) {
        const int n   = blockN + waveN * 64 + (j << 4) + l15;
        const float bv = bias[n];
        v8f acc = (j == 0) ? acc0 : (j == 1) ? acc1 : (j == 2) ? acc2 : acc3;
        #pragma unroll
        for (int v = 0; v < 8; ++v) {
            const int m = m0 + v;
            float val = acc[v] + bv;
            if (addX) val += X[(size_t)m * N + n];
            if (relu) val = fmaxf(val, 0.0f);
            C[(size_t)m * N + n] = val;
        }
    }
}

static inline void launch_gemm(const float* A, const float* W, const float* X,
                               const float* bias, float* C,
                               int M, int N, int K,
                               int useA, int addX, int relu, hipStream_t stream)
{
    dim3 grid(N / 128, M / 64);
    rnn_gemm_wmma_f32<<<grid, 256, 0, stream>>>(A, W, X, bias, C,
                                                M, N, K, useA, addX, relu);
}

extern "C" void kernel_launch(void* const* d_in, const int* in_sizes, int n_in,
                              void* d_out, int out_size, void* d_ws, size_t ws_size,
                              hipStream_t stream)
{
    (void)in_sizes; (void)n_in; (void)out_size; (void)ws_size;

    const float* x     = (const float*)d_in[0];   // [512, 128]
    const float* W_x2h = (const float*)d_in[1];   // [512, 128]
    const float* b_x2h = (const float*)d_in[2];   // [512]
    const float* W_h2h = (const float*)d_in[3];   // [512, 512]
    const float* b_h2h = (const float*)d_in[4];   // [512]
    const float* W_h2y = (const float*)d_in[5];   // [256, 512]
    const float* b_h2y = (const float*)d_in[6];   // [256]
    float* out = (float*)d_out;                   // [512, 256]

    const int B = 512, DIN = 128, H = 512, DOUT = 256, T = 128;

    float* XH = (float*)d_ws;                     // [512, 512]
    float* h0 = XH + (size_t)B * H;               // [512, 512]
    float* h1 = h0 + (size_t)B * H;               // [512, 512]

    // Phase 1: XH = x @ W_x2h^T + b_x2h   (no ReLU)
    launch_gemm(x, W_x2h, XH /*unused*/, b_x2h, XH,
                B, H, DIN, /*useA=*/1, /*addX=*/0, /*relu=*/0, stream);

    // Step 0: h = relu(XH + b_h2h)  (h starts at zero => matmul skipped)
    launch_gemm(XH /*unused*/, W_h2h, XH, b_h2h, h0,
                B, H, H, /*useA=*/0, /*addX=*/1, /*relu=*/1, stream);

    // Steps 1..T-1: h = relu(XH + h @ W_h2h^T + b_h2h), double buffered.
    float* cur = h0;
    for (int t = 1; t < T; ++t) {
        float* nxt = (cur == h0) ? h1 : h0;
        launch_gemm(cur, W_h2h, XH, b_h2h, nxt,
                    B, H, H, /*useA=*/1, /*addX=*/1, /*relu=*/1, stream);
        cur = nxt;
    }

    // Phase 3: out = h @ W_h2y^T + b_h2y
    launch_gemm(cur, W_h2y, XH /*unused*/, b_h2y, out,
                B, DOUT, H, /*useA=*/1, /*addX=*/0, /*relu=*/0, stream);
}